// LGP_32538672235156
// MI455X (gfx1250) — compile-verified
//
#include <hip/hip_runtime.h>

#define Npts 6144
#define Kn   16
#define NK   (Npts*Kn)
#define EPSB 1e-5f

typedef float v2f __attribute__((ext_vector_type(2)));
typedef float v8f __attribute__((ext_vector_type(8)));

union V8 { v8f v; float f[8]; };

// ---------------- float atomic max via CAS (portable) -------------------
__device__ inline void atomicMaxF(float* addr, float val) {
    int* ai = (int*)addr;
    int cur = *ai;
    while (__int_as_float(cur) < val) {
        int assumed = cur;
        cur = atomicCAS(ai, assumed, __float_as_int(val));
        if (cur == assumed) break;
    }
}

// ---------------- K0: init scatter buffers + BN stats -------------------
__global__ void lgp_init(float* smax, float* den, float* resid, float* bn) {
    int idx = blockIdx.x * blockDim.x + threadIdx.x;
    const int NS = Npts * 48;
    if (idx < NS)            smax[idx]          = -3.0e38f;
    else if (idx < 2 * NS)   den[idx - NS]      = 0.f;
    else if (idx < 3 * NS)   resid[idx - 2*NS]  = 0.f;
    else if (idx < 3*NS+384) bn[idx - 3*NS]     = 0.f;
}

// ---------------- K1: shrink/xv + atomic segment max --------------------
__global__ void lgp_shrink(const float* __restrict__ p, const float* __restrict__ x,
                           const int* __restrict__ knn,
                           const float* __restrict__ Wpos, const float* __restrict__ bpos,
                           const float* __restrict__ Wlin, const float* __restrict__ blin,
                           const float* __restrict__ Wlinx, const float* __restrict__ blinx,
                           float* shrink, float* xv, float* smax) {
    __shared__ float peL[4][64];
    __shared__ float xfL[4][64];
    const int t = threadIdx.x;
    const int g = t >> 6, c = t & 63;
    const int r = blockIdx.x * 4 + g;
    const int i = r >> 4;
    const int j = knn[r];
    const float r0 = p[j*3+0] - p[i*3+0];
    const float r1 = p[j*3+1] - p[i*3+1];
    const float r2 = p[j*3+2] - p[i*3+2];
    xfL[g][c] = x[j*64 + c];
    peL[g][c] = bpos[c] + r0*Wpos[c] + r1*Wpos[64+c] + r2*Wpos[128+c];
    __syncthreads();
    if (c < 48) {
        float s = blin[c], sx = blinx[c];
        for (int c2 = 0; c2 < 64; ++c2) {
            s  += peL[g][c2] * Wlin[c2*48 + c];
            s  += xfL[g][c2] * Wlin[(64 + c2)*48 + c];
            sx += xfL[g][c2] * Wlinx[c2*48 + c];
        }
        shrink[r*48 + c] = s;
        xv[r*48 + c]     = sx;
        atomicMaxF(&smax[j*48 + c], s);
    }
}

// ---------------- K2: exp + segment denominator -------------------------
__global__ void lgp_exp(const int* __restrict__ knn, float* shrink,
                        const float* __restrict__ smax, float* den) {
    int idx = blockIdx.x * blockDim.x + threadIdx.x;
    if (idx >= NK * 48) return;
    const int r = idx / 48, c = idx % 48;
    const int j = knn[r];
    float e = expf(shrink[idx] - smax[j*48 + c]);
    shrink[idx] = e;
    atomicAdd(&den[j*48 + c], e);
}

// ---------------- K3: prob * xv scatter-sum -----------------------------
__global__ void lgp_resid(const int* __restrict__ knn, const float* __restrict__ shrink,
                          const float* __restrict__ xv, const float* __restrict__ den,
                          float* resid) {
    int idx = blockIdx.x * blockDim.x + threadIdx.x;
    if (idx >= NK * 48) return;
    const int r = idx / 48, c = idx % 48;
    const int j = knn[r];
    float prob = shrink[idx] / den[j*48 + c];
    atomicAdd(&resid[j*48 + c], xv[idx] * prob);
}

// ---------------- K4: pos -> h (first linears) + BN partials ------------
__global__ void lgp_h(const float* __restrict__ p, const int* __restrict__ knn,
                      const float* __restrict__ resid,
                      const float* __restrict__ Wq1, const float* __restrict__ bq1,
                      const float* __restrict__ Wk1, const float* __restrict__ bk1,
                      const float* __restrict__ Wv1, const float* __restrict__ bv1,
                      float* hq, float* hk, float* hv, float* bn) {
    __shared__ float kxL[48], resL[48], posL[16*48], bnpL[192], bnp2L[192];
    const int i = blockIdx.x, t = threadIdx.x;
    if (t < 48) {
        int a = t / 3, t3 = t % 3;
        kxL[t]  = p[knn[i*16 + a]*3 + t3];
        resL[t] = resid[i*48 + t];
    }
    if (t < 192) { bnpL[t] = 0.f; bnp2L[t] = 0.f; }
    __syncthreads();
#pragma unroll
    for (int e = 0; e < 3; ++e) {
        int idx = t + 256*e;
        if (idx < 768) {
            int a = idx / 48, s = idx % 48, b = s / 3, t3 = s % 3;
            posL[idx] = kxL[b*3 + t3] - kxL[a*3 + t3] + resL[s];
        }
    }
    __syncthreads();
    const float* W1s[3] = {Wq1, Wk1, Wv1};
    const float* b1s[3] = {bq1, bk1, bv1};
    float* hb[3] = {hq, hk, hv};
#pragma unroll
    for (int e = 0; e < 12; ++e) {
        int o = t + 256*e;                 // 3072 outputs
        int br = o >> 10, rem = o & 1023, a = rem >> 6, c = rem & 63;
        const float* W1 = W1s[br];
        float hval = b1s[br][c];
        const float* pr = &posL[a*48];
        for (int s = 0; s < 48; ++s) hval += pr[s] * W1[s*64 + c];
        hb[br][(i*16 + a)*64 + c] = hval;
        atomicAdd(&bnpL[br*64 + c], hval);
        atomicAdd(&bnp2L[br*64 + c], hval * hval);
    }
    __syncthreads();
    if (t < 192)      atomicAdd(&bn[t], bnpL[t]);
    else if (t < 384) atomicAdd(&bn[t], bnp2L[t - 192]);
}

// ---------------- K5: finalize BN stats ---------------------------------
__global__ void lgp_bn(const float* __restrict__ bn, float* bnfin) {
    int t = threadIdx.x;
    if (t < 192) {
        float mean = bn[t] / (float)NK;
        float var  = bn[192 + t] / (float)NK - mean * mean;
        bnfin[t]       = mean;
        bnfin[192 + t] = rsqrtf(var + EPSB);
    }
}

// ---------------- K6: normalize + relu + second linear (in-place) -------
__global__ void lgp_posmlp(float* hq, float* hk, float* hv,
                           const float* __restrict__ bnfin,
                           const float* __restrict__ gq, const float* __restrict__ beq,
                           const float* __restrict__ Wq2, const float* __restrict__ bq2,
                           const float* __restrict__ gk, const float* __restrict__ bek,
                           const float* __restrict__ Wk2, const float* __restrict__ bk2,
                           const float* __restrict__ gv, const float* __restrict__ bev,
                           const float* __restrict__ Wv2, const float* __restrict__ bv2) {
    __shared__ float nhL[4][64];
    const int t = threadIdx.x, g = t >> 6, c = t & 63;
    const int r = blockIdx.x * 4 + g;
    float* hb[3] = {hq, hk, hv};
    const float* gs[3]  = {gq, gk, gv};
    const float* bes[3] = {beq, bek, bev};
    const float* W2s[3] = {Wq2, Wk2, Wv2};
    const float* b2s[3] = {bq2, bk2, bv2};
    for (int br = 0; br < 3; ++br) {
        float h  = hb[br][r*64 + c];
        float nh = (h - bnfin[br*64 + c]) * bnfin[192 + br*64 + c] * gs[br][c] + bes[br][c];
        nh = nh > 0.f ? nh : 0.f;
        nhL[g][c] = nh;
        __syncthreads();
        float o = b2s[br][c];
        const float* W2 = W2s[br];
        for (int c2 = 0; c2 < 64; ++c2) o += nhL[g][c2] * W2[c2*64 + c];
        hb[br][r*64 + c] = o;
        __syncthreads();
    }
}

// ---------------- K7: fused per-point attention (WMMA, all-LDS) ---------
// LDS ~225 KB -> 1 block/WGP, so VGPR budget is effectively unconstrained:
// register-block B fragments (W_pg/W_pp) per wave and share each A fragment
// between the two WMMA accumulators.
__global__ void __launch_bounds__(256) lgp_attn(
    const float* __restrict__ p, const float* __restrict__ x, const int* __restrict__ knn,
    const float* __restrict__ Wqkv,
    const float* __restrict__ Wpos, const float* __restrict__ bpos,
    const float* __restrict__ Wpg, const float* __restrict__ bpg,
    const float* __restrict__ Wpp, const float* __restrict__ bpp,
    const float* __restrict__ Wproj, const float* __restrict__ bproj,
    const float* __restrict__ posq, const float* __restrict__ posk,
    const float* __restrict__ posv, float* out) {
    extern __shared__ float smem[];
    float* kx      = smem;                 // 64   (48 used)
    float* xk      = kx + 64;              // 16x64
    float* pqL     = xk + 1024;            // 16x64
    float* pkL     = pqL + 1024;
    float* pvL     = pkL + 1024;
    float* qkvL    = pvL + 1024;           // 16x192
    float* attnL   = qkvL + 3072;          // 4x16x16
    float* scoresL = attnL + 1024;         // 4x16
    float* outcL   = scoresL + 64;         // 64
    float* pembL   = outcL + 64;           // 256x64
    float* g4L     = pembL + 16384;        // 256x64
    float* pp4L    = g4L + 16384;          // 256x64

    const int i = blockIdx.x;
    const int t = threadIdx.x;
    const int lane = t & 31, wave = t >> 5;

    if (t < 48) { int a = t / 3, t3 = t % 3; kx[t] = p[knn[i*16 + a]*3 + t3]; }
#pragma unroll
    for (int e = 0; e < 4; ++e) {
        int idx = t + 256*e;
        int a = idx >> 6, c = idx & 63;
        int src = (i*16 + a)*64 + c;
        xk[idx]  = x[knn[i*16 + a]*64 + c];
        pqL[idx] = posq[src];
        pkL[idx] = posk[src];
        pvL[idx] = posv[src];
    }
    __syncthreads();

    { // pemb4[256][64] = rel_pos @ W_pos + b_pos   (K=3: VALU)
        int a = t >> 4, b = t & 15;
        float r0 = kx[b*3]   - kx[a*3];
        float r1 = kx[b*3+1] - kx[a*3+1];
        float r2 = kx[b*3+2] - kx[a*3+2];
        float* dst = &pembL[t*64];
        for (int c = 0; c < 64; ++c)
            dst[c] = bpos[c] + r0*Wpos[c] + r1*Wpos[64+c] + r2*Wpos[128+c];
    }
    __syncthreads();

    const int kh = (lane >> 4) * 2;      // A/B frag K-offset: lanes 16-31 -> K2/K3
    const int l15 = lane & 15;
    const int ro = (lane >> 4) * 8;      // C/D frag row offset

    // ---- qkv[16][192] = x_knn @ W_qkv (WMMA fp32; A fragments hoisted) ----
    {
        v2f Af[16];
#pragma unroll
        for (int ks = 0; ks < 16; ++ks) {
            int k0 = ks * 4 + kh;
            v2f tmp = { xk[l15*64 + k0], xk[l15*64 + k0 + 1] };
            Af[ks] = tmp;
        }
        for (int nt = wave; nt < 12; nt += 8) {
            const int n = nt*16 + l15;
            V8 acc;
#pragma unroll
            for (int r = 0; r < 8; ++r) acc.f[r] = 0.f;
#pragma unroll
            for (int ks = 0; ks < 16; ++ks) {
                int k0 = ks * 4 + kh;
                v2f bf = { Wqkv[k0*192 + n], Wqkv[(k0+1)*192 + n] };
                acc.v = __builtin_amdgcn_wmma_f32_16x16x4_f32(
                    false, Af[ks], false, bf, (short)0, acc.v, false, false);
            }
#pragma unroll
            for (int r = 0; r < 8; ++r) qkvL[(r + ro)*192 + n] = acc.f[r];
        }
    }

    // ---- g4/pp4[256][64] = pemb4 @ {W_pg,W_pp} + bias --------------------
    // Each wave owns one N-tile (nt = wave&3); 2 waves split the 16 M-tiles.
    // B fragments for both weights preloaded once (64 VGPRs), A fragment
    // loaded once per k-step and fed to BOTH accumulators.
    {
        const int nt  = wave & 3;
        const int mt0 = (wave >> 2) * 8;
        const int n   = nt*16 + l15;
        v2f Bg[16], Bp[16];
#pragma unroll
        for (int ks = 0; ks < 16; ++ks) {
            int k0 = ks * 4 + kh;
            v2f tg = { Wpg[k0*64 + n], Wpg[(k0+1)*64 + n] };
            v2f tp = { Wpp[k0*64 + n], Wpp[(k0+1)*64 + n] };
            Bg[ks] = tg; Bp[ks] = tp;
        }
        const float biasg = bpg[n];
        const float biasp = bpp[n];
        for (int mi = 0; mi < 8; ++mi) {
            const int mt = mt0 + mi;
            const int m  = mt*16 + l15;
            V8 a1, a2;
#pragma unroll
            for (int r = 0; r < 8; ++r) { a1.f[r] = 0.f; a2.f[r] = 0.f; }
#pragma unroll
            for (int ks = 0; ks < 16; ++ks) {
                int k0 = ks * 4 + kh;
                v2f af = { pembL[m*64 + k0], pembL[m*64 + k0 + 1] };
                a1.v = __builtin_amdgcn_wmma_f32_16x16x4_f32(
                    false, af, false, Bg[ks], (short)0, a1.v, false, false);
                a2.v = __builtin_amdgcn_wmma_f32_16x16x4_f32(
                    false, af, false, Bp[ks], (short)0, a2.v, false, false);
            }
#pragma unroll
            for (int r = 0; r < 8; ++r) {
                g4L[(mt*16 + r + ro)*64 + n]  = a1.f[r] + biasg;
                pp4L[(mt*16 + r + ro)*64 + n] = a2.f[r] + biasp;
            }
        }
    }
    __syncthreads();

    // attention rows: dots = q.kT + q.pqT + k.pkT ; softmax(*0.25)
    if (t < 64) {
        int h = t >> 4, a = t & 15;
        const float* qr = &qkvL[a*192 + h*16];
        const float* kr = &qkvL[a*192 + 64 + h*16];
        float dv[16]; float mx = -3.0e38f;
        for (int j = 0; j < 16; ++j) {
            const float* kj  = &qkvL[j*192 + 64 + h*16];
            const float* pqj = &pqL[j*64 + h*16];
            const float* pkj = &pkL[j*64 + h*16];
            float s = 0.f;
            for (int d = 0; d < 16; ++d) s += qr[d]*kj[d] + qr[d]*pqj[d] + kr[d]*pkj[d];
            s *= 0.25f;
            dv[j] = s; if (s > mx) mx = s;
        }
        float den = 0.f;
        for (int j = 0; j < 16; ++j) { float e = expf(dv[j] - mx); dv[j] = e; den += e; }
        float inv = 1.f / den;
        for (int j = 0; j < 16; ++j) attnL[(h*16 + a)*16 + j] = dv[j] * inv;
    }
    __syncthreads();

    // pos_states output: [n,h,16,16]
    const size_t OUT1 = (size_t)Npts * 64;
#pragma unroll
    for (int e = 0; e < 4; ++e) {
        int idx = t + 256*e;
        int h = idx >> 8, rem = idx & 255, a = rem >> 4, d = rem & 15;
        const float* ar = &attnL[(h*16 + a)*16];
        float s = 0.f;
        for (int j = 0; j < 16; ++j) s += ar[j] * g4L[(a*16 + j)*64 + h*16 + d];
        out[OUT1 + ((((size_t)i*4 + h)*16 + a)*16 + d)] = s;
    }
    // scores (only k=0 row of `out` is consumed by reference)
    if (t < 64) {
        int h = t >> 4, j = t & 15;
        float sp = 0.f;
        for (int d = 0; d < 16; ++d) sp += qkvL[h*16 + d] * pp4L[j*64 + h*16 + d];
        scoresL[t] = (attnL[(h*16)*16 + j] + sp) * 0.5f;   // /sqrt(H)=2
    }
    __syncthreads();
    if (t < 64) {
        int h = t >> 4, d = t & 15;
        float s = 0.f;
        for (int j = 0; j < 16; ++j)
            s += scoresL[h*16 + j] * (qkvL[j*192 + 128 + h*16 + d] + pvL[j*64 + h*16 + d]);
        outcL[t] = s;
    }
    __syncthreads();
    if (t < 64) {
        float s = bproj[t];
        for (int c2 = 0; c2 < 64; ++c2) s += outcL[c2] * Wproj[c2*64 + t];
        out[(size_t)i*64 + t] = s;
    }
}

// ------------------------------------------------------------------------
extern "C" void kernel_launch(void* const* d_in, const int* in_sizes, int n_in,
                              void* d_out, int out_size, void* d_ws, size_t ws_size,
                              hipStream_t stream) {
    const float* p     = (const float*)d_in[0];
    const float* x     = (const float*)d_in[1];
    const int*   knn   = (const int*)  d_in[2];
    const float* Wqkv  = (const float*)d_in[3];
    const float* Wproj = (const float*)d_in[4];
    const float* bproj = (const float*)d_in[5];
    const float* Wpos  = (const float*)d_in[6];
    const float* bpos  = (const float*)d_in[7];
    const float* Wlin  = (const float*)d_in[8];
    const float* blin  = (const float*)d_in[9];
    const float* Wlinx = (const float*)d_in[10];
    const float* blinx = (const float*)d_in[11];
    const float* Wpp   = (const float*)d_in[12];
    const float* bpp   = (const float*)d_in[13];
    const float* Wpg   = (const float*)d_in[14];
    const float* bpg   = (const float*)d_in[15];
    const float* Wq1 = (const float*)d_in[16];
    const float* bq1 = (const float*)d_in[17];
    const float* gq  = (const float*)d_in[18];
    const float* beq = (const float*)d_in[19];
    const float* Wq2 = (const float*)d_in[20];
    const float* bq2 = (const float*)d_in[21];
    const float* Wk1 = (const float*)d_in[22];
    const float* bk1 = (const float*)d_in[23];
    const float* gk  = (const float*)d_in[24];
    const float* bek = (const float*)d_in[25];
    const float* Wk2 = (const float*)d_in[26];
    const float* bk2 = (const float*)d_in[27];
    const float* Wv1 = (const float*)d_in[28];
    const float* bv1 = (const float*)d_in[29];
    const float* gv  = (const float*)d_in[30];
    const float* bev = (const float*)d_in[31];
    const float* Wv2 = (const float*)d_in[32];
    const float* bv2 = (const float*)d_in[33];
    float* out = (float*)d_out;

    float* ws = (float*)d_ws;
    size_t o = 0;
    float* shrink = ws + o; o += (size_t)NK * 48;
    float* xv     = ws + o; o += (size_t)NK * 48;
    float* smax   = ws + o; o += (size_t)Npts * 48;
    float* den    = ws + o; o += (size_t)Npts * 48;
    float* resid  = ws + o; o += (size_t)Npts * 48;
    float* hq     = ws + o; o += (size_t)NK * 64;
    float* hk     = ws + o; o += (size_t)NK * 64;
    float* hv     = ws + o; o += (size_t)NK * 64;
    float* bn     = ws + o; o += 384;
    float* bnfin  = ws + o; o += 384;

    const int initN = 3 * Npts * 48 + 384;
    lgp_init  <<<(initN + 255)/256, 256, 0, stream>>>(smax, den, resid, bn);
    lgp_shrink<<<NK/4, 256, 0, stream>>>(p, x, knn, Wpos, bpos, Wlin, blin,
                                         Wlinx, blinx, shrink, xv, smax);
    lgp_exp   <<<(NK*48 + 255)/256, 256, 0, stream>>>(knn, shrink, smax, den);
    lgp_resid <<<(NK*48 + 255)/256, 256, 0, stream>>>(knn, shrink, xv, den, resid);
    lgp_h     <<<Npts, 256, 0, stream>>>(p, knn, resid, Wq1, bq1, Wk1, bk1,
                                         Wv1, bv1, hq, hk, hv, bn);
    lgp_bn    <<<1, 256, 0, stream>>>(bn, bnfin);
    lgp_posmlp<<<NK/4, 256, 0, stream>>>(hq, hk, hv, bnfin,
                                         gq, beq, Wq2, bq2,
                                         gk, bek, Wk2, bk2,
                                         gv, bev, Wv2, bv2);
    const size_t shmem = (size_t)(64 + 4*1024 + 3072 + 1024 + 64 + 64 + 3*16384)
                         * sizeof(float);   // ~225 KB, fits 320 KB/WGP LDS
    lgp_attn  <<<Npts, 256, shmem, stream>>>(p, x, knn, Wqkv, Wpos, bpos,
                                             Wpg, bpg, Wpp, bpp, Wproj, bproj,
                                             hq, hk, hv, out);
}